// WACConv_11450382811893
// MI455X (gfx1250) — compile-verified
//
#include <hip/hip_runtime.h>
#include <math.h>

#define D_DIM 200
#define LDA   204          // LDS stride: 12 banks apart per row; conflict-free (gcd(12,64)=4)
#define NTILES 13          // ceil(200/16): last tile cols 192..207, clamped loads
#define MBLK  64           // rows per workgroup (4 waves x 16 rows)

typedef __attribute__((ext_vector_type(2))) float v2f;
typedef __attribute__((ext_vector_type(8))) float v8f;

__device__ __forceinline__ float gelu_f(float x) {
    float x3 = x * x * x;
    return 0.5f * x * (1.0f + tanhf(0.7978845608f * (x + 0.044715f * x3)));
}

// ---------------------------------------------------------------------------
// degree: deg[row] += 1 for each message-passing edge
// ---------------------------------------------------------------------------
__global__ void degree_kernel(const int* __restrict__ ei, float* __restrict__ deg, int num_e) {
    int e = blockIdx.x * blockDim.x + threadIdx.x;
    if (e < num_e) {
        int row = ei[num_e + e];          // out_index[0] = edge_index[0, num_e + e]
        atomicAdd(&deg[row], 1.0f);
    }
}

__global__ void deginv_kernel(const float* __restrict__ deg, float* __restrict__ dinv, int N) {
    int i = blockIdx.x * blockDim.x + threadIdx.x;
    if (i < N) {
        float d = deg[i];
        dinv[i] = (d > 0.0f) ? (1.0f / sqrtf(d)) : 0.0f;
    }
}

// ---------------------------------------------------------------------------
// edge messages (GEMM hoisted past the segment_sum):
//   pre[row] += norm * x[col] * (rel_e*ew + rw)
// One float4 chunk (4 dims) per work-item; 50 chunks per edge. The 80 MB
// accumulator is L2-resident (192 MB L2), so the atomics resolve at L2.
// ---------------------------------------------------------------------------
__global__ __launch_bounds__(256) void edge_msg_kernel(
    const int* __restrict__ ei, const int* __restrict__ et,
    const float* __restrict__ x, const float* __restrict__ rel_embed,
    const float* __restrict__ lr, const float* __restrict__ rw1,
    const float* __restrict__ ew1, const float* __restrict__ dinv,
    float* __restrict__ pre, int num_e, int nrel2)
{
    long idx = (long)blockIdx.x * blockDim.x + threadIdx.x;
    long total = (long)num_e * (D_DIM / 4);
    if (idx >= total) return;
    int e = (int)(idx / (D_DIM / 4));
    int q = (int)(idx - (long)e * (D_DIM / 4));

    int row = ei[num_e + e];              // out_index[0]
    int col = ei[3 * num_e + e];          // out_index[1]
    int in0 = ei[e];                      // in_index[0] (first half!)
    int t   = et[num_e + e];              // out_type

    float norm = dinv[row] * dinv[col];

    const float4* xj = (const float4*)(x + (long)col * D_DIM);
    const float4* re = (t < nrel2) ? (const float4*)(rel_embed + (long)t * D_DIM)
                                   : (const float4*)lr;   // rel_all last row = loop_rel
    const float4* rw = (const float4*)(rw1 + (long)t * D_DIM);
    const float4* ew = (const float4*)(ew1 + (long)in0 * D_DIM);

    float4 xv = xj[q], rv = re[q], wv = rw[q], ev = ew[q];
    float* dst = pre + (long)row * D_DIM + q * 4;
    atomicAdd(dst + 0, xv.x * (rv.x * ev.x + wv.x) * norm);
    atomicAdd(dst + 1, xv.y * (rv.y * ev.y + wv.y) * norm);
    atomicAdd(dst + 2, xv.z * (rv.z * ev.z + wv.z) * norm);
    atomicAdd(dst + 3, xv.w * (rv.w * ev.w + wv.w) * norm);
}

// ---------------------------------------------------------------------------
// Fused WMMA GEMM (M x 200) @ (200 x 200), fp32 V_WMMA_F32_16X16X4_F32.
// Workgroup = 4 waves = 64-row M block; A staged in LDS once, B staged per
// 16-column N-tile (transposed, stride-204 -> bank-conflict-free b64 reads).
// No predication anywhere in the MAC loop: out-of-range columns are clamped
// on load and masked only at store time, so EXEC is all-ones at every WMMA.
//   mode 0: A = x * ent_weight2 * loop_rel   (selfloop path)
//   mode 1: A = pre                          (aggregated messages; in-place OK)
//   mode 2: A = rel_all (rel_embed + loop_rel row)
// If attn != null, also computes a[m] = sum_col gelu(acc)*attn[col] per row
// via in-register accumulation + half-wave shfl_xor reduction.
// ---------------------------------------------------------------------------
__global__ __launch_bounds__(128) void gemm_wmma_kernel(
    const float* __restrict__ A0, const float* __restrict__ A1,
    const float* __restrict__ lr, const float* __restrict__ B,
    float* __restrict__ outp, float* __restrict__ arow,
    const float* __restrict__ attn,
    int Mtotal, int Mstore, int mode, float scale, int doGelu)
{
    __shared__ float As[MBLK * LDA];   // 52.2 KB
    __shared__ float Bt[16 * LDA];     // 13.1 KB  (total 63.75 KB)

    const int m0  = blockIdx.x * MBLK;
    const int tid = threadIdx.x;

    // cooperative A-block load (64 x 200) with per-mode transform
    for (int idx = tid; idx < MBLK * D_DIM; idx += 128) {
        int r = idx / D_DIM, k = idx - r * D_DIM;
        int mg = m0 + r;
        float v = 0.0f;
        if (mg < Mtotal) {
            if (mode == 0)      v = A0[(long)mg * D_DIM + k] * A1[(long)mg * D_DIM + k] * lr[k];
            else if (mode == 1) v = A0[(long)mg * D_DIM + k];
            else                v = (mg < Mtotal - 1) ? A0[(long)mg * D_DIM + k] : lr[k];
        }
        As[r * LDA + k] = v;
    }

    const int lane = tid & 31;
    const int wv   = tid >> 5;     // wave id 0..3 -> owns rows m0+16*wv ..+15
    const int lg   = lane >> 4;    // 0 -> K+{0,1}, 1 -> K+{2,3}
    const int ln   = lane & 15;

    const float* Arow = &As[(wv * 16 + ln) * LDA + 2 * lg];

    float apart[8];
    #pragma unroll
    for (int v = 0; v < 8; ++v) apart[v] = 0.0f;

    for (int t = 0; t < NTILES; ++t) {
        const int n0 = t * 16;
        __syncthreads();                       // Bt free from previous tile
        // stage B panel transposed: Bt[n_local][k]; clamp col (no predication)
        for (int idx = tid; idx < 16 * D_DIM; idx += 128) {
            int nl = idx & 15, k = idx >> 4;
            int colc = n0 + nl; colc = (colc < D_DIM) ? colc : (D_DIM - 1);
            Bt[nl * LDA + k] = B[k * D_DIM + colc];
        }
        __syncthreads();

        // prefetch next tile's B panel into cache while we compute
        if (t + 1 < NTILES) {
            int nn0 = (t + 1) * 16; nn0 = (nn0 < D_DIM - 16) ? nn0 : (D_DIM - 16);
            for (int k = tid; k < D_DIM; k += 128)
                __builtin_prefetch(&B[k * D_DIM + nn0], 0, 0);
        }

        const float* Brow = &Bt[ln * LDA + 2 * lg];
        v8f acc = {};
        #pragma unroll 5
        for (int k0 = 0; k0 < D_DIM; k0 += 4) {
            v2f a, b;
            a.x = Arow[k0]; a.y = Arow[k0 + 1];
            b.x = Brow[k0]; b.y = Brow[k0 + 1];
            acc = __builtin_amdgcn_wmma_f32_16x16x4_f32(false, a, false, b,
                                                        (short)0, acc, false, false);
        }

        const int col = n0 + ln;
        const bool cok = (col < D_DIM);
        const float am = (attn != nullptr && cok) ? attn[col] : 0.0f;
        #pragma unroll
        for (int v = 0; v < 8; ++v) {          // C: VGPR v -> M = v + 8*lg
            int mg = m0 + wv * 16 + 8 * lg + v;
            float val = acc[v];
            if (doGelu) val = gelu_f(val);
            apart[v] += val * am;
            if (cok && mg < Mstore) outp[(long)mg * D_DIM + col] = val * scale;
        }
    }

    if (arow != nullptr) {
        #pragma unroll
        for (int v = 0; v < 8; ++v) {
            float s = apart[v];
            s += __shfl_xor(s, 1, 32);         // reduce within each 16-lane group
            s += __shfl_xor(s, 2, 32);
            s += __shfl_xor(s, 4, 32);
            s += __shfl_xor(s, 8, 32);
            if (ln == 0) {
                int mg = m0 + wv * 16 + 8 * lg + v;
                if (mg < Mstore) arow[mg] = s;
            }
        }
    }
}

// ---------------------------------------------------------------------------
// combine: out = 0.75*s1 (already in out) + 0.25 * a[n] * g2[n]
// plus first stage of batchnorm column sums (per-block LDS partials).
// ---------------------------------------------------------------------------
__global__ __launch_bounds__(256) void combine_kernel(
    const float* __restrict__ g2, const float* __restrict__ arow,
    float* __restrict__ outp, float* __restrict__ csum, float* __restrict__ csumsq, int N)
{
    __shared__ float cs[D_DIM];
    __shared__ float cq[D_DIM];
    const int n0 = blockIdx.x * 16;
    const int tid = threadIdx.x;

    for (int d = tid; d < D_DIM; d += 256) { cs[d] = 0.0f; cq[d] = 0.0f; }
    __syncthreads();

    for (int idx = tid; idx < 16 * D_DIM; idx += 256) {
        int r = idx / D_DIM, d = idx - r * D_DIM;
        int n = n0 + r;
        if (n < N) {
            long off = (long)n * D_DIM + d;
            float o = outp[off] + 0.25f * arow[n] * g2[off];
            outp[off] = o;
            atomicAdd(&cs[d], o);
            atomicAdd(&cq[d], o * o);
        }
    }
    __syncthreads();

    for (int d = tid; d < D_DIM; d += 256) {
        atomicAdd(&csum[d],   cs[d]);
        atomicAdd(&csumsq[d], cq[d]);
    }
}

// ---------------------------------------------------------------------------
// batchnorm finalize (biased variance, matches jnp.var)
// ---------------------------------------------------------------------------
__global__ void bn_kernel(float* __restrict__ outp,
                          const float* __restrict__ csum, const float* __restrict__ csumsq,
                          const float* __restrict__ gamma, const float* __restrict__ beta, int N)
{
    long idx = (long)blockIdx.x * blockDim.x + threadIdx.x;
    long total = (long)N * D_DIM;
    if (idx >= total) return;
    int d = (int)(idx % D_DIM);
    float invN = 1.0f / (float)N;
    float mean = csum[d] * invN;
    float var  = csumsq[d] * invN - mean * mean;
    float o = outp[idx];
    outp[idx] = (o - mean) * rsqrtf(var + 1e-5f) * gamma[d] + beta[d];
}

// ---------------------------------------------------------------------------
extern "C" void kernel_launch(void* const* d_in, const int* in_sizes, int n_in,
                              void* d_out, int out_size, void* d_ws, size_t ws_size,
                              hipStream_t stream)
{
    const float* x         = (const float*)d_in[0];
    const float* rel_embed = (const float*)d_in[1];
    const int*   ei        = (const int*)  d_in[2];
    const int*   et        = (const int*)  d_in[3];
    const float* w1_loop   = (const float*)d_in[4];
    const float* w1_out    = (const float*)d_in[5];
    const float* w_rel     = (const float*)d_in[6];
    const float* lr        = (const float*)d_in[7];   // loop_rel (1 x D)
    const float* attn      = (const float*)d_in[8];   // attn_w   (1 x D)
    const float* rw1       = (const float*)d_in[9];   // rel_weight1 (475 x D)
    const float* ew1       = (const float*)d_in[10];  // ent_weight1 (N x D)
    const float* ew2       = (const float*)d_in[11];  // ent_weight2 (N x D)
    const float* gamma     = (const float*)d_in[12];
    const float* beta      = (const float*)d_in[13];

    const int N     = in_sizes[0] / D_DIM;      // 100000
    const int nrel2 = in_sizes[1] / D_DIM;      // 474
    const int num_e = in_sizes[3] / 2;          // 400000

    float* out     = (float*)d_out;                       // N x D
    float* out_rel = out + (long)N * D_DIM;               // 474 x D

    // workspace layout (~81 MB): pre is reused in-place as g2 after GEMM2
    float* pre    = (float*)d_ws;                         // N x D accumulator -> g2
    float* deg    = pre + (long)N * D_DIM;                // N
    float* dinv   = deg + N;                              // N
    float* csum   = dinv + N;                             // D
    float* csumsq = csum + D_DIM;                         // D
    float* arow   = csumsq + D_DIM;                       // N (attention dots)

    hipMemsetAsync(pre,  0, (size_t)N * D_DIM * sizeof(float), stream);
    hipMemsetAsync(deg,  0, (size_t)N * sizeof(float), stream);
    hipMemsetAsync(csum, 0, 2 * D_DIM * sizeof(float), stream);

    degree_kernel<<<(num_e + 255) / 256, 256, 0, stream>>>(ei, deg, num_e);
    deginv_kernel<<<(N + 255) / 256, 256, 0, stream>>>(deg, dinv, N);

    {
        long items = (long)num_e * (D_DIM / 4);
        edge_msg_kernel<<<(int)((items + 255) / 256), 256, 0, stream>>>(
            ei, et, x, rel_embed, lr, rw1, ew1, dinv, pre, num_e, nrel2);
    }

    const int mblocksN = (N + MBLK - 1) / MBLK;
    // selfloop: out = 0.75 * gelu((x*ew2*lr) @ w1_loop)
    gemm_wmma_kernel<<<mblocksN, 128, 0, stream>>>(x, ew2, lr, w1_loop, out,
                                                   nullptr, nullptr, N, N, 0, 0.75f, 1);
    // aggregated messages: g2 = gelu(pre @ w1_out) in-place over pre,
    // fused attention row-dot -> arow
    gemm_wmma_kernel<<<mblocksN, 128, 0, stream>>>(pre, nullptr, lr, w1_out, pre,
                                                   arow, attn, N, N, 1, 1.0f, 1);
    // relation1 = (rel_all @ w_rel)[:-1]
    const int Mrel = nrel2 + 1;
    gemm_wmma_kernel<<<(Mrel + MBLK - 1) / MBLK, 128, 0, stream>>>(
        rel_embed, nullptr, lr, w_rel, out_rel, nullptr, nullptr, Mrel, nrel2, 2, 1.0f, 0);

    combine_kernel<<<(N + 15) / 16, 256, 0, stream>>>(pre, arow, out, csum, csumsq, N);
    bn_kernel<<<(int)(((long)N * D_DIM + 255) / 256), 256, 0, stream>>>(
        out, csum, csumsq, gamma, beta, N);
}